// GraphClassifier_28398323761927
// MI455X (gfx1250) — compile-verified
//
#include <hip/hip_runtime.h>

typedef float v2f __attribute__((ext_vector_type(2)));
typedef float v8f __attribute__((ext_vector_type(8)));

#define NNODES   100000
#define HIDCH    64
#define NCLASSES 10
#define NGRAPHS  64

// ---------------------------------------------------------------------------
// zero fill
// ---------------------------------------------------------------------------
__global__ void k_zero(float* __restrict__ p, int n) {
    int i = blockIdx.x * blockDim.x + threadIdx.x;
    if (i < n) p[i] = 0.0f;
}

// ---------------------------------------------------------------------------
// Y[r][o] = sum_c X[r][c] * W[o][c]   (X:[nrows,64], W:[64,64] row-major)
// One wave32 per 16-row tile; fp32 WMMA 16x16x4, 16 K-steps x 4 N-tiles.
// A-layout: lane&15 = M row; lanes>=16 hold K+2/K+3 halves.
// ---------------------------------------------------------------------------
__global__ __launch_bounds__(128)
void k_gemm(const float* __restrict__ X, const float* __restrict__ W,
            float* __restrict__ Y, int nrows) {
    int wave = blockIdx.x * (blockDim.x >> 5) + (threadIdx.x >> 5);
    int lane = threadIdx.x & 31;
    int row0 = wave << 4;
    if (row0 >= nrows) return;                 // wave-uniform exit: EXEC stays full

    int m     = lane & 15;
    int khalf = (lane >> 4) << 1;              // 0 or 2
    int rbase = row0 + ((lane >> 4) << 3);     // D-layout: M = v + 8*(lane>=16)

    const float* xrow = X + (size_t)(row0 + m) * HIDCH;
    float ax[16], ay[16];
    #pragma unroll
    for (int kk = 0; kk < 16; ++kk) {          // each lane holds half the row
        ax[kk] = xrow[kk * 4 + khalf];
        ay[kk] = xrow[kk * 4 + khalf + 1];
    }

    #pragma unroll
    for (int ot = 0; ot < 4; ++ot) {
        v8f c = {0.f, 0.f, 0.f, 0.f, 0.f, 0.f, 0.f, 0.f};
        const float* wrow = W + (size_t)(ot * 16 + (lane & 15)) * HIDCH;
        #pragma unroll
        for (int kk = 0; kk < 16; ++kk) {
            v2f a; a.x = ax[kk];                 a.y = ay[kk];
            v2f b; b.x = wrow[kk * 4 + khalf];   b.y = wrow[kk * 4 + khalf + 1];
            c = __builtin_amdgcn_wmma_f32_16x16x4_f32(false, a, false, b,
                                                      (short)0, c, false, false);
        }
        int col = ot * 16 + (lane & 15);
        #pragma unroll
        for (int v = 0; v < 8; ++v)
            Y[(size_t)(rbase + v) * HIDCH + col] = c[v];
    }
}

// ---------------------------------------------------------------------------
// H[r][o] = relu( AGG[r][o] + sum_c X[r][c]*W[o][c] + bias[o] )
// AGG tile is preloaded straight into the WMMA C operand.
// ---------------------------------------------------------------------------
__global__ __launch_bounds__(128)
void k_combine(const float* __restrict__ X, const float* __restrict__ W,
               const float* __restrict__ bias, const float* __restrict__ AGG,
               float* __restrict__ H, int nrows) {
    int wave = blockIdx.x * (blockDim.x >> 5) + (threadIdx.x >> 5);
    int lane = threadIdx.x & 31;
    int row0 = wave << 4;
    if (row0 >= nrows) return;

    int m     = lane & 15;
    int khalf = (lane >> 4) << 1;
    int rbase = row0 + ((lane >> 4) << 3);

    const float* xrow = X + (size_t)(row0 + m) * HIDCH;
    float ax[16], ay[16];
    #pragma unroll
    for (int kk = 0; kk < 16; ++kk) {
        ax[kk] = xrow[kk * 4 + khalf];
        ay[kk] = xrow[kk * 4 + khalf + 1];
    }

    #pragma unroll
    for (int ot = 0; ot < 4; ++ot) {
        int col = ot * 16 + (lane & 15);
        v8f c;
        #pragma unroll
        for (int v = 0; v < 8; ++v)            // C = aggregated neighbor features
            c[v] = AGG[(size_t)(rbase + v) * HIDCH + col];

        const float* wrow = W + (size_t)(ot * 16 + (lane & 15)) * HIDCH;
        #pragma unroll
        for (int kk = 0; kk < 16; ++kk) {
            v2f a; a.x = ax[kk];                 a.y = ay[kk];
            v2f b; b.x = wrow[kk * 4 + khalf];   b.y = wrow[kk * 4 + khalf + 1];
            c = __builtin_amdgcn_wmma_f32_16x16x4_f32(false, a, false, b,
                                                      (short)0, c, false, false);
        }
        float bv = bias[col];
        #pragma unroll
        for (int v = 0; v < 8; ++v) {
            float r = c[v] + bv;
            H[(size_t)(rbase + v) * HIDCH + col] = r > 0.f ? r : 0.f;
        }
    }
}

// ---------------------------------------------------------------------------
// agg[dst[e]] += rel[src[e]]  :  16 threads per edge, float4 rows, f32 atomics
// ---------------------------------------------------------------------------
__global__ void k_scatter(const long long* __restrict__ src,
                          const long long* __restrict__ dst,
                          const float* __restrict__ rel,
                          float* __restrict__ agg, int nedges) {
    int t = blockIdx.x * blockDim.x + threadIdx.x;
    if (t >= nedges * 16) return;
    int e = t >> 4;
    int q = t & 15;
    long long s = src[e];
    long long d = dst[e];
    const float4 v = *(const float4*)(rel + (size_t)s * HIDCH + q * 4);
    float* ap = agg + (size_t)d * HIDCH + q * 4;
    atomicAdd(ap + 0, v.x);
    atomicAdd(ap + 1, v.y);
    atomicAdd(ap + 2, v.z);
    atomicAdd(ap + 3, v.w);
}

// ---------------------------------------------------------------------------
// global mean-pool accumulation: sums[g] += h[node], counts[g] += 1
// ---------------------------------------------------------------------------
__global__ void k_pool(const float* __restrict__ h, const long long* __restrict__ batch,
                       float* __restrict__ sums, float* __restrict__ counts, int nnodes) {
    int t = blockIdx.x * blockDim.x + threadIdx.x;
    if (t >= nnodes * HIDCH) return;
    int node = t >> 6;
    int ch   = t & 63;
    int g    = (int)batch[node];
    atomicAdd(&sums[g * HIDCH + ch], h[t]);
    if (ch == 0) atomicAdd(&counts[g], 1.0f);
}

// ---------------------------------------------------------------------------
// out[g][k] = (sums[g]/max(cnt,1)) . W_lin[k] + b_lin[k]   -> [64,10]
// ---------------------------------------------------------------------------
__global__ void k_head(const float* __restrict__ sums, const float* __restrict__ counts,
                       const float* __restrict__ Wl, const float* __restrict__ bl,
                       float* __restrict__ out) {
    int idx = blockIdx.x * blockDim.x + threadIdx.x;
    if (idx >= NGRAPHS * NCLASSES) return;
    int g = idx / NCLASSES;
    int k = idx % NCLASSES;
    float cnt = counts[g];
    cnt = cnt > 1.0f ? cnt : 1.0f;
    float inv = 1.0f / cnt;
    float acc = 0.f;
    #pragma unroll
    for (int c = 0; c < HIDCH; ++c)
        acc += (sums[g * HIDCH + c] * inv) * Wl[k * HIDCH + c];
    out[idx] = acc + bl[k];
}

// ---------------------------------------------------------------------------
extern "C" void kernel_launch(void* const* d_in, const int* in_sizes, int n_in,
                              void* d_out, int out_size, void* d_ws, size_t ws_size,
                              hipStream_t stream) {
    const float*     x      = (const float*)d_in[0];
    const long long* ei     = (const long long*)d_in[1];
    const long long* batch  = (const long long*)d_in[2];
    const float*     W1rel  = (const float*)d_in[3];
    const float*     W1root = (const float*)d_in[4];
    const float*     b1     = (const float*)d_in[5];
    const float*     W2rel  = (const float*)d_in[6];
    const float*     W2root = (const float*)d_in[7];
    const float*     b2     = (const float*)d_in[8];
    const float*     Wlin   = (const float*)d_in[9];
    const float*     blin   = (const float*)d_in[10];
    float*           out    = (float*)d_out;

    const int E  = in_sizes[1] / 2;
    const long long* src = ei;
    const long long* dst = ei + E;

    const int NF = NNODES * HIDCH;              // 6.4M floats per node buffer
    float* A      = (float*)d_ws;
    float* B      = A + NF;
    float* C      = B + NF;
    float* pooled = C + NF;                     // 64*64
    float* counts = pooled + NGRAPHS * HIDCH;   // 64

    const int zb    = (NF + 255) / 256;
    const int tiles = NNODES / 16;              // 6250, exact
    const int gb    = (tiles + 3) / 4;          // 4 waves / block
    const int sb    = (E * 16 + 255) / 256;

    // ---- layer 1 ----
    k_zero   <<<zb, 256, 0, stream>>>(B, NF);                      // agg1 = 0
    k_gemm   <<<gb, 128, 0, stream>>>(x, W1rel, A, NNODES);        // A = x @ W1_rel^T
    k_scatter<<<sb, 256, 0, stream>>>(src, dst, A, B, E);          // B = segsum(A[src])
    k_combine<<<gb, 128, 0, stream>>>(x, W1root, b1, B, C, NNODES);// C = h1

    // ---- layer 2 ----
    k_zero   <<<zb, 256, 0, stream>>>(A, NF);                      // agg2 = 0
    k_gemm   <<<gb, 128, 0, stream>>>(C, W2rel, B, NNODES);        // B = h1 @ W2_rel^T
    k_scatter<<<sb, 256, 0, stream>>>(src, dst, B, A, E);          // A = segsum(B[src])
    k_combine<<<gb, 128, 0, stream>>>(C, W2root, b2, A, B, NNODES);// B = h2

    // ---- mean pool + head ----
    k_zero<<<(NGRAPHS * HIDCH + NGRAPHS + 255) / 256, 256, 0, stream>>>(pooled,
                                                    NGRAPHS * HIDCH + NGRAPHS);
    k_pool<<<(NF + 255) / 256, 256, 0, stream>>>(B, batch, pooled, counts, NNODES);
    k_head<<<(NGRAPHS * NCLASSES + 255) / 256, 256, 0, stream>>>(pooled, counts,
                                                                 Wlin, blin, out);
}